// EntanglementGNN_18906446037215
// MI455X (gfx1250) — compile-verified
//
#include <hip/hip_runtime.h>
#include <hip/hip_bf16.h>

#define N_NODES 50000
#define N_EDGES 500000
#define H 128
#define LAYERS 4
#define NODE_IN 7
#define EDGE_IN 4
#define OUT_DIM 8
#define LN_EPS 1e-5f

typedef __attribute__((ext_vector_type(2))) float v2f;
typedef __attribute__((ext_vector_type(4))) float v4f;
typedef __attribute__((ext_vector_type(8))) float v8f;

// D = A(16x4 f32) * B(4x16 f32) + C(16x16 f32), fp32 matrix op on CDNA5.
__device__ __forceinline__ v8f wmma4(v2f a, v2f b, v8f c) {
  return __builtin_amdgcn_wmma_f32_16x16x4_f32(false, a, false, b, (short)0, c,
                                               false, false);
}

__device__ __forceinline__ v8f v8f_zero() {
  v8f z = {0.f, 0.f, 0.f, 0.f, 0.f, 0.f, 0.f, 0.f};
  return z;
}

// ---------------- encoder: h = node_x @ enc_w (7x128) + b ----------------
__global__ void k_encode_nodes(const float* __restrict__ x,
                               const float* __restrict__ w,
                               const float* __restrict__ b,
                               float* __restrict__ h) {
  int idx = blockIdx.x * blockDim.x + threadIdx.x;
  if (idx >= N_NODES * H) return;
  int n = idx >> 7, c = idx & (H - 1);
  float acc = b[c];
#pragma unroll
  for (int k = 0; k < NODE_IN; ++k) acc += x[n * NODE_IN + k] * w[k * H + c];
  h[idx] = acc;
}

// ---------------- in-degree counts ----------------
__global__ void k_counts(const int* __restrict__ tgt, float* __restrict__ counts) {
  int e = blockIdx.x * blockDim.x + threadIdx.x;
  if (e < N_EDGES) unsafeAtomicAdd(&counts[tgt[e]], 1.0f);
}

// ------- message + scatter: one wave per 32-edge tile (2 M-subtiles) --------
// Each B-fragment (shared weights) feeds two WMMAs -> halves L2 weight traffic.
#define MI_S 388  // 384 + 4 pad -> rows land on distinct LDS banks for A-frags
#define EM 32     // edges per wave

__global__ void __launch_bounds__(32)
k_msg(const float* __restrict__ h, float* __restrict__ agg,
      const float* __restrict__ et, const int* __restrict__ src,
      const int* __restrict__ tgt,
      const float* __restrict__ ecw, const float* __restrict__ ecb,
      const float* __restrict__ w1, const float* __restrict__ b1,
      const float* __restrict__ w2, const float* __restrict__ b2) {
  __shared__ float mi[EM * MI_S];  // [32 edges][384] concat(h_src, h_tgt, enc(e))
  const int ebase = blockIdx.x * EM;
  const int lane = threadIdx.x;
  const int ln = lane & 15;         // col-in-tile (B/C) or row-in-subtile (A)
  const int kh = (lane >> 4) << 1;  // K sub-offset: 0 or 2
  const int rh = (lane >> 4) << 3;  // C row offset: 0 or 8
  const int c4 = lane * 4;

  // Stage mi tile: coalesced 512B row loads + fused edge encoder (K=4).
  for (int r = 0; r < EM; ++r) {
    int s = src[ebase + r];
    int t = tgt[ebase + r];
    *(v4f*)&mi[r * MI_S + c4] = *(const v4f*)&h[(size_t)s * H + c4];
    *(v4f*)&mi[r * MI_S + 128 + c4] = *(const v4f*)&h[(size_t)t * H + c4];
    float e0 = et[(ebase + r) * EDGE_IN + 0];
    float e1 = et[(ebase + r) * EDGE_IN + 1];
    float e2 = et[(ebase + r) * EDGE_IN + 2];
    float e3 = et[(ebase + r) * EDGE_IN + 3];
    v4f ev;
#pragma unroll
    for (int j = 0; j < 4; ++j) {
      int c = c4 + j;
      ev[j] = ecb[c] + e0 * ecw[0 * H + c] + e1 * ecw[1 * H + c] +
              e2 * ecw[2 * H + c] + e3 * ecw[3 * H + c];
    }
    *(v4f*)&mi[r * MI_S + 256 + c4] = ev;
  }

  // Stage 1: m1 = relu(mi[32,384] @ w1[384,128] + b1); all 16 accumulators
  // (8 N-tiles x 2 M-subtiles) held in VGPRs until mi can be overlaid.
  v8f acc[8][2];
#pragma unroll
  for (int nt = 0; nt < 8; ++nt) {
    acc[nt][0] = v8f_zero();
    acc[nt][1] = v8f_zero();
  }
#pragma unroll
  for (int nt = 0; nt < 8; ++nt) {
    const int n = nt * 16 + ln;
#pragma unroll 4
    for (int kb = 0; kb < 3 * H; kb += 4) {
      v2f b;
      b.x = w1[(kb + kh) * H + n];
      b.y = w1[(kb + kh + 1) * H + n];
      v2f a0 = *(const v2f*)&mi[ln * MI_S + kb + kh];
      v2f a1 = *(const v2f*)&mi[(16 + ln) * MI_S + kb + kh];
      acc[nt][0] = wmma4(a0, b, acc[nt][0]);
      acc[nt][1] = wmma4(a1, b, acc[nt][1]);
    }
  }
  // ReLU + bias; overlay m1 into mi cols 0..127 (all stage-1 reads are done).
#pragma unroll
  for (int nt = 0; nt < 8; ++nt) {
    const int n = nt * 16 + ln;
    float bias = b1[n];
#pragma unroll
    for (int i = 0; i < 8; ++i) {
      float v0 = acc[nt][0][i] + bias;
      float v1 = acc[nt][1][i] + bias;
      mi[(i + rh) * MI_S + n] = v0 > 0.f ? v0 : 0.f;
      mi[(16 + i + rh) * MI_S + n] = v1 > 0.f ? v1 : 0.f;
    }
  }

  // Stage 2: m = m1[32,128] @ w2[128,128] + b2; scatter-add into agg[tgt].
#pragma unroll
  for (int nt = 0; nt < 8; ++nt) {
    const int n = nt * 16 + ln;
    v8f a20 = v8f_zero();
    v8f a21 = v8f_zero();
#pragma unroll 4
    for (int kb = 0; kb < H; kb += 4) {
      v2f b;
      b.x = w2[(kb + kh) * H + n];
      b.y = w2[(kb + kh + 1) * H + n];
      v2f a0 = *(const v2f*)&mi[ln * MI_S + kb + kh];
      v2f a1 = *(const v2f*)&mi[(16 + ln) * MI_S + kb + kh];
      a20 = wmma4(a0, b, a20);
      a21 = wmma4(a1, b, a21);
    }
    float bias = b2[n];
#pragma unroll
    for (int i = 0; i < 8; ++i) {
      int t0 = tgt[ebase + i + rh];
      unsafeAtomicAdd(&agg[(size_t)t0 * H + n], a20[i] + bias);
      int t1 = tgt[ebase + 16 + i + rh];
      unsafeAtomicAdd(&agg[(size_t)t1 * H + n], a21[i] + bias);
    }
  }
}

// ---------------- node update + residual + layernorm (one wave / 16 nodes) ----
#define UI_S 260  // 256 + 4 pad

__global__ void __launch_bounds__(32)
k_upd(float* __restrict__ h, const float* __restrict__ agg,
      const float* __restrict__ counts,
      const float* __restrict__ w1, const float* __restrict__ b1,
      const float* __restrict__ w2, const float* __restrict__ b2,
      const float* __restrict__ lng, const float* __restrict__ lnb) {
  __shared__ float ui[16 * UI_S];  // [16][256] = [h | agg/cnt]; later overlaid
  const int nbase = blockIdx.x * 16;
  const int lane = threadIdx.x;
  const int ln = lane & 15;
  const int kh = (lane >> 4) << 1;
  const int rh = (lane >> 4) << 3;
  const int c4 = lane * 4;

  for (int r = 0; r < 16; ++r) {
    int node = nbase + r;
    float cnt = counts[node];
    cnt = cnt > 1.f ? cnt : 1.f;
    *(v4f*)&ui[r * UI_S + c4] = *(const v4f*)&h[(size_t)node * H + c4];
    v4f av = *(const v4f*)&agg[(size_t)node * H + c4];
    *(v4f*)&ui[r * UI_S + 128 + c4] = av * (1.0f / cnt);
  }

  // Stage 1: accumulate all 8 N-tiles before overwriting ui with u1.
  v8f accs[8];
#pragma unroll
  for (int nt = 0; nt < 8; ++nt) {
    const int n = nt * 16 + ln;
    v8f acc = v8f_zero();
#pragma unroll 8
    for (int kb = 0; kb < 2 * H; kb += 4) {
      v2f a = *(const v2f*)&ui[ln * UI_S + kb + kh];
      v2f b;
      b.x = w1[(kb + kh) * H + n];
      b.y = w1[(kb + kh + 1) * H + n];
      acc = wmma4(a, b, acc);
    }
    accs[nt] = acc;
  }
#pragma unroll
  for (int nt = 0; nt < 8; ++nt) {
    const int n = nt * 16 + ln;
    float bias = b1[n];
#pragma unroll
    for (int i = 0; i < 8; ++i) {
      float v = accs[nt][i] + bias;
      ui[(i + rh) * UI_S + n] = v > 0.f ? v : 0.f;  // u1 in cols 0..127
    }
  }

  // Stage 2: u = u1 @ w2 + b2; v = h + u stored to cols 128..255 for LN.
#pragma unroll
  for (int nt = 0; nt < 8; ++nt) {
    const int n = nt * 16 + ln;
    v8f acc = v8f_zero();
#pragma unroll 8
    for (int kb = 0; kb < H; kb += 4) {
      v2f a = *(const v2f*)&ui[ln * UI_S + kb + kh];
      v2f b;
      b.x = w2[(kb + kh) * H + n];
      b.y = w2[(kb + kh + 1) * H + n];
      acc = wmma4(a, b, acc);
    }
    float bias = b2[n];
#pragma unroll
    for (int i = 0; i < 8; ++i) {
      int row = i + rh;
      float v = acc[i] + bias + h[(size_t)(nbase + row) * H + n];
      ui[row * UI_S + 128 + n] = v;
    }
  }

  // LayerNorm: 2 lanes per row, combine halves with a cross-half shuffle.
  const int half = lane >> 4;
  float s1 = 0.f, s2 = 0.f;
  for (int j = 0; j < 64; ++j) {
    float v = ui[ln * UI_S + 128 + half * 64 + j];
    s1 += v;
    s2 += v * v;
  }
  s1 += __shfl_xor(s1, 16, 32);
  s2 += __shfl_xor(s2, 16, 32);
  float mean = s1 * (1.f / 128.f);
  float var = s2 * (1.f / 128.f) - mean * mean;
  float rs = rsqrtf(var + LN_EPS);
  int node = nbase + ln;
  for (int j = 0; j < 64; ++j) {
    int c = half * 64 + j;
    float v = ui[ln * UI_S + 128 + c];
    h[(size_t)node * H + c] = (v - mean) * rs * lng[c] + lnb[c];
  }
}

// ---------------- node output head (one wave / 16 nodes) ----------------
#define HT_S 132

__global__ void __launch_bounds__(32)
k_out(const float* __restrict__ h,
      const float* __restrict__ w1, const float* __restrict__ b1,
      const float* __restrict__ w2, const float* __restrict__ b2,
      float* __restrict__ out) {
  __shared__ float t[16 * HT_S];
  const int nbase = blockIdx.x * 16;
  const int lane = threadIdx.x;
  const int ln = lane & 15;
  const int kh = (lane >> 4) << 1;
  const int rh = (lane >> 4) << 3;
  const int c4 = lane * 4;

  for (int r = 0; r < 16; ++r)
    *(v4f*)&t[r * HT_S + c4] = *(const v4f*)&h[(size_t)(nbase + r) * H + c4];

  v8f accs[8];
#pragma unroll
  for (int nt = 0; nt < 8; ++nt) {
    const int n = nt * 16 + ln;
    v8f acc = v8f_zero();
#pragma unroll 8
    for (int kb = 0; kb < H; kb += 4) {
      v2f a = *(const v2f*)&t[ln * HT_S + kb + kh];
      v2f b;
      b.x = w1[(kb + kh) * H + n];
      b.y = w1[(kb + kh + 1) * H + n];
      acc = wmma4(a, b, acc);
    }
    accs[nt] = acc;
  }
#pragma unroll
  for (int nt = 0; nt < 8; ++nt) {
    const int n = nt * 16 + ln;
    float bias = b1[n];
#pragma unroll
    for (int i = 0; i < 8; ++i) {
      float v = accs[nt][i] + bias;
      t[(i + rh) * HT_S + n] = v > 0.f ? v : 0.f;
    }
  }

  // Stage 2: [16,128] @ w2[128,8] with N padded to 16 (cols 8..15 zeroed).
  v8f acc = v8f_zero();
#pragma unroll 8
  for (int kb = 0; kb < H; kb += 4) {
    v2f a = *(const v2f*)&t[ln * HT_S + kb + kh];
    v2f b;
    b.x = (ln < OUT_DIM) ? w2[(kb + kh) * OUT_DIM + ln] : 0.f;
    b.y = (ln < OUT_DIM) ? w2[(kb + kh + 1) * OUT_DIM + ln] : 0.f;
    acc = wmma4(a, b, acc);
  }
  if (ln < OUT_DIM) {
    float bias = b2[ln];
#pragma unroll
    for (int i = 0; i < 8; ++i)
      out[(size_t)(nbase + i + rh) * OUT_DIM + ln] = acc[i] + bias;
  }
}

// ---------------- graph mean + head ----------------
__global__ void k_gsum(const float* __restrict__ h, float* __restrict__ gsum) {
  int c = threadIdx.x;  // 128 threads, 200 blocks x 250 nodes
  int nb = blockIdx.x * 250;
  float s = 0.f;
  for (int r = 0; r < 250; ++r) s += h[(size_t)(nb + r) * H + c];
  unsafeAtomicAdd(&gsum[c], s);
}

__global__ void k_ghead(const float* __restrict__ gsum,
                        const float* __restrict__ w1, const float* __restrict__ b1,
                        const float* __restrict__ w2, const float* __restrict__ b2,
                        float* __restrict__ out) {
  __shared__ float g[H];
  __shared__ float u1[H];
  int c = threadIdx.x;
  g[c] = gsum[c] * (1.0f / (float)N_NODES);
  __syncthreads();
  float a = b1[c];
  for (int k = 0; k < H; ++k) a += g[k] * w1[k * H + c];
  u1[c] = a > 0.f ? a : 0.f;
  __syncthreads();
  if (c < OUT_DIM) {
    float o = b2[c];
    for (int k = 0; k < H; ++k) o += u1[k] * w2[k * OUT_DIM + c];
    out[(size_t)N_NODES * OUT_DIM + c] = o;
  }
}

extern "C" void kernel_launch(void* const* d_in, const int* in_sizes, int n_in,
                              void* d_out, int out_size, void* d_ws, size_t ws_size,
                              hipStream_t stream) {
  const float* node_x = (const float*)d_in[0];
  const float* edge_x = (const float*)d_in[1];
  const int* eidx = (const int*)d_in[2];
  const float* enw = (const float*)d_in[3];
  const float* enb = (const float*)d_in[4];
  const float* eew = (const float*)d_in[5];
  const float* eeb = (const float*)d_in[6];
  const float* mw1 = (const float*)d_in[7];
  const float* mb1 = (const float*)d_in[8];
  const float* mw2 = (const float*)d_in[9];
  const float* mb2 = (const float*)d_in[10];
  const float* uw1 = (const float*)d_in[11];
  const float* ub1 = (const float*)d_in[12];
  const float* uw2 = (const float*)d_in[13];
  const float* ub2 = (const float*)d_in[14];
  const float* lng = (const float*)d_in[15];
  const float* lnb = (const float*)d_in[16];
  const float* ow1 = (const float*)d_in[17];
  const float* ob1 = (const float*)d_in[18];
  const float* ow2 = (const float*)d_in[19];
  const float* ob2 = (const float*)d_in[20];
  const float* gw1 = (const float*)d_in[21];
  const float* gb1 = (const float*)d_in[22];
  const float* gw2 = (const float*)d_in[23];
  const float* gb2 = (const float*)d_in[24];
  float* out = (float*)d_out;

  const int* src = eidx;
  const int* tgt = eidx + N_EDGES;

  char* ws = (char*)d_ws;
  float* h = (float*)(ws);                                     // N*H
  float* agg = (float*)(ws + (size_t)N_NODES * H * 4);         // N*H
  float* counts = (float*)(ws + (size_t)2 * N_NODES * H * 4);  // N
  float* gsum = counts + N_NODES;                              // H

  hipMemsetAsync(counts, 0, (N_NODES + H) * sizeof(float), stream);
  k_encode_nodes<<<(N_NODES * H + 255) / 256, 256, 0, stream>>>(node_x, enw, enb, h);
  k_counts<<<(N_EDGES + 255) / 256, 256, 0, stream>>>(tgt, counts);

  for (int l = 0; l < LAYERS; ++l) {
    hipMemsetAsync(agg, 0, (size_t)N_NODES * H * sizeof(float), stream);
    k_msg<<<N_EDGES / EM, 32, 0, stream>>>(
        h, agg, edge_x, src, tgt, eew, eeb,
        mw1 + (size_t)l * 3 * H * H, mb1 + (size_t)l * H,
        mw2 + (size_t)l * H * H, mb2 + (size_t)l * H);
    k_upd<<<N_NODES / 16, 32, 0, stream>>>(
        h, agg, counts,
        uw1 + (size_t)l * 2 * H * H, ub1 + (size_t)l * H,
        uw2 + (size_t)l * H * H, ub2 + (size_t)l * H,
        lng + (size_t)l * H, lnb + (size_t)l * H);
  }

  k_out<<<N_NODES / 16, 32, 0, stream>>>(h, ow1, ob1, ow2, ob2, out);
  k_gsum<<<200, 128, 0, stream>>>(h, gsum);
  k_ghead<<<1, 128, 0, stream>>>(gsum, gw1, gb1, gw2, gb2, out);
}